// DefDETRPostProcessor_82506321756353
// MI455X (gfx1250) — compile-verified
//
#include <hip/hip_runtime.h>
#include <math.h>

#define NB 1024          // batch
#define NQ 300           // queries
#define NC 91            // classes
#define NN (NQ * NC)     // 27300 flattened scores per batch row
#define NK 100           // top-k
#define NT 256           // threads per block (8 waves of 32)

// CDNA5 async global->LDS path (gfx1250). Guarded so we compile either way.
#if defined(__gfx1250__) && __has_builtin(__builtin_amdgcn_global_load_async_to_lds_b128)
#define USE_ASYNC_LDS 1
#endif

typedef int v4i __attribute__((ext_vector_type(4)));
typedef __attribute__((address_space(1))) v4i glb_v4i;   // global (AS1) int4
typedef __attribute__((address_space(3))) v4i lds_v4i;   // LDS (AS3) int4

// Order-preserving float -> uint transform (ascending uint == ascending float)
__device__ __forceinline__ unsigned int mono_key(float f) {
    unsigned int u = __float_as_uint(f);
    return (u & 0x80000000u) ? ~u : (u | 0x80000000u);
}
__device__ __forceinline__ float mono_inv(unsigned int k) {
    unsigned int u = (k & 0x80000000u) ? (k & 0x7fffffffu) : ~k;
    return __uint_as_float(u);
}

// 64-bit select key: value-desc, then lowest flat index first (matches lax.top_k)
#define KEY64(kf, i) ((((unsigned long long)(kf)) << 32) | (unsigned int)(~(unsigned int)(i)))

__global__ __launch_bounds__(NT) void detr_post_kernel(
    const float* __restrict__ logits,   // [NB, NQ, NC]
    const float* __restrict__ pboxes,   // [NB, NQ, 4] (cx,cy,w,h)
    const int*   __restrict__ tsizes,   // [NB, 2] (h, w)
    float*       __restrict__ out)      // scores | labels | boxes (flat concat)
{
    extern __shared__ __align__(16) unsigned int smem[];
    unsigned int* keys = smem;            // NN   monotonic keys of logits
    unsigned int* hist = smem + NN;       // 256  radix histogram
    unsigned int* sel  = hist + 256;      // NK   winning flat indices
    unsigned int* ctl  = sel + NK;        // 8    [0..1]=prefix [2]=remaining [3]=counter [4]=done

    const int b   = blockIdx.x;
    const int tid = threadIdx.x;
    const float* gl = logits + (size_t)b * NN;

    // ---- Stage 1: HBM -> LDS, one pass. NN = 27300 = 6825 float4's ----
#ifdef USE_ASYNC_LDS
    for (int i = tid; i < NN / 4; i += NT) {
        __builtin_amdgcn_global_load_async_to_lds_b128(
            (glb_v4i*)(gl + 4 * i), (lds_v4i*)(keys + 4 * i),
            /*offset=*/0, /*cpol=*/0);
    }
#if __has_builtin(__builtin_amdgcn_s_wait_asynccnt)
    __builtin_amdgcn_s_wait_asynccnt(0);
#else
    asm volatile("s_wait_asynccnt 0" ::: "memory");
#endif
#else
    for (int i = tid; i < NN / 4; i += NT) {
        reinterpret_cast<float4*>(keys)[i] = reinterpret_cast<const float4*>(gl)[i];
    }
#endif
    __syncthreads();

    // ---- Stage 2: in-place monotonic key transform (sigmoid is monotone) ----
    for (int i4 = tid; i4 < NN / 4; i4 += NT) {
        uint4 kv = reinterpret_cast<uint4*>(keys)[i4];
        uint4 o;
        o.x = mono_key(__uint_as_float(kv.x));
        o.y = mono_key(__uint_as_float(kv.y));
        o.z = mono_key(__uint_as_float(kv.z));
        o.w = mono_key(__uint_as_float(kv.w));
        reinterpret_cast<uint4*>(keys)[i4] = o;
    }
    __syncthreads();

    // ---- Stage 3: radix select (MSB-first, early exit) for the NK-th largest key ----
    unsigned long long prefix = 0ull;
    unsigned int remaining = NK;
    unsigned long long kthr = 0ull;
    bool have_thr = false;
    for (int pass = 0; pass < 8; ++pass) {
        const int shift = 56 - 8 * pass;
        hist[tid] = 0;               // NT == 256
        __syncthreads();
        for (int i4 = tid; i4 < NN / 4; i4 += NT) {
            const uint4 kv = reinterpret_cast<const uint4*>(keys)[i4];
            const unsigned int ka[4] = {kv.x, kv.y, kv.z, kv.w};
            const int base = 4 * i4;
#pragma unroll
            for (int c = 0; c < 4; ++c) {
                unsigned long long k = KEY64(ka[c], base + c);
                bool in_pref = (pass == 0) || ((k >> (shift + 8)) == prefix);
                if (in_pref)
                    atomicAdd(&hist[(unsigned int)(k >> shift) & 255u], 1u);
            }
        }
        __syncthreads();
        if (tid == 0) {
            unsigned int cum = 0;
            int bin = 255;
            for (; bin > 0; --bin) {
                unsigned int c = hist[bin];
                if (cum + c >= remaining) break;
                cum += c;
            }
            prefix = (prefix << 8) | (unsigned int)bin;
            remaining -= cum;
            ctl[0] = (unsigned int)(prefix >> 32);
            ctl[1] = (unsigned int)prefix;
            ctl[2] = remaining;
            ctl[3] = 0;              // reset compaction counter
            // If the selected bin holds exactly `remaining` elements, every
            // element with (k >> shift) >= prefix is a winner: done.
            ctl[4] = (hist[bin] == remaining) ? 1u : 0u;
        }
        __syncthreads();
        prefix    = ((unsigned long long)ctl[0] << 32) | ctl[1];
        remaining = ctl[2];
        if (ctl[4] != 0u) {
            kthr = prefix << shift;   // shift in [0,56]
            have_thr = true;
            break;
        }
    }
    if (!have_thr) kthr = prefix;     // full 64-bit key after 8 passes (distinct keys)

    // ---- Stage 4: compact the winners (exactly NK hits by construction) ----
    for (int i4 = tid; i4 < NN / 4; i4 += NT) {
        const uint4 kv = reinterpret_cast<const uint4*>(keys)[i4];
        const unsigned int ka[4] = {kv.x, kv.y, kv.z, kv.w};
        const int base = 4 * i4;
#pragma unroll
        for (int c = 0; c < 4; ++c) {
            unsigned long long k = KEY64(ka[c], base + c);
            if (k >= kthr) {
                unsigned int pos = atomicAdd(&ctl[3], 1u);
                if (pos < NK) sel[pos] = (unsigned int)(base + c);
            }
        }
    }
    __syncthreads();

    // ---- Stage 5: rank by counting (distinct keys -> unique ranks), emit ----
    if (tid < NK) {
        const unsigned int i = sel[tid];
        const unsigned long long kt = KEY64(keys[i], i);
        int rank = 0;
        for (int j = 0; j < NK; ++j) {
            unsigned int ij = sel[j];
            rank += (KEY64(keys[ij], ij) > kt) ? 1 : 0;
        }

        const int query = (int)(i / NC);
        const int label = (int)(i % NC);
        const float logit = mono_inv(keys[i]);
        const float score = 1.0f / (1.0f + expf(-logit));

        const float* gb = pboxes + ((size_t)b * NQ + query) * 4;
        const float cx = gb[0], cy = gb[1], w = gb[2], h = gb[3];
        const float img_h = (float)tsizes[b * 2 + 0];
        const float img_w = (float)tsizes[b * 2 + 1];
        const float hw = 0.5f * w, hh = 0.5f * h;

        float* scores_o = out;
        float* labels_o = out + (size_t)NB * NK;
        float* boxes_o  = out + (size_t)2 * NB * NK;

        const size_t o = (size_t)b * NK + (size_t)rank;
        scores_o[o] = score;
        labels_o[o] = (float)label;
        float* bo = boxes_o + o * 4;
        bo[0] = (cx - hw) * img_w;
        bo[1] = (cy - hh) * img_h;
        bo[2] = (cx + hw) * img_w;
        bo[3] = (cy + hh) * img_h;
    }
}

extern "C" void kernel_launch(void* const* d_in, const int* in_sizes, int n_in,
                              void* d_out, int out_size, void* d_ws, size_t ws_size,
                              hipStream_t stream) {
    const float* logits = (const float*)d_in[0];
    const float* boxes  = (const float*)d_in[1];
    const int*   tsz    = (const int*)d_in[2];
    float* out = (float*)d_out;

    const size_t smem_bytes = (size_t)(NN + 256 + NK + 8) * sizeof(unsigned int); // ~110.7 KB
    (void)hipFuncSetAttribute((const void*)detr_post_kernel,
                              hipFuncAttributeMaxDynamicSharedMemorySize,
                              (int)smem_bytes);
    detr_post_kernel<<<NB, NT, smem_bytes, stream>>>(logits, boxes, tsz, out);
}